// RegionProposalNetwork_18159121727680
// MI455X (gfx1250) — compile-verified
//
#include <hip/hip_runtime.h>
#include <stdint.h>

#define N_IMG     16
#define A_TOTAL   261888
#define K_TOT     4768          // 1000+1000+1000+1000+768, = 298*16
#define KW        149           // ceil(4768/32) bitmap words per row
#define NTILE     298
#define SORT_N    8192
#define NMS_THR   0.7f
#define POST_NMS  1000
#define IMG_SZ    1024.0f
#define BBOX_CLAMP 4.135166556742356f   // log(1000/16)

typedef __attribute__((ext_vector_type(2))) float v2f;
typedef __attribute__((ext_vector_type(8))) float v8f;

__device__ __forceinline__ unsigned mono_key(float f) {
  unsigned u = __float_as_uint(f);
  return u ^ ((unsigned)((int)u >> 31) | 0x80000000u);
}

// ---------------- Kernel 1: per (image, level) radix top-k select ----------
__global__ void topk_kernel(const float* __restrict__ objn, int* __restrict__ sel) {
  __shared__ unsigned hist[256];
  __shared__ unsigned sh_prefix, sh_krem;
  __shared__ int sh_cgt, sh_ceq;
  const int LN[5] = {196608, 49152, 12288, 3072, 768};
  const int LO[5] = {0, 196608, 245760, 258048, 261120};
  int img = blockIdx.x, lvl = blockIdx.y;
  int n = LN[lvl], base = LO[lvl];
  int k = (lvl < 4) ? 1000 : 768;
  const float* ob = objn + (size_t)img * A_TOTAL + base;
  int tid = threadIdx.x, nth = blockDim.x;
  if (tid == 0) { sh_prefix = 0u; sh_krem = (unsigned)k; }
  __syncthreads();
  for (int pass = 0; pass < 4; ++pass) {
    int shift = 24 - 8 * pass;
    for (int b = tid; b < 256; b += nth) hist[b] = 0u;
    __syncthreads();
    unsigned pmask = pass ? (0xFFFFFFFFu << (shift + 8)) : 0u;
    unsigned prefix = sh_prefix;
    for (int i = tid; i < n; i += nth) {
      unsigned key = mono_key(ob[i]);
      if ((key & pmask) == (prefix & pmask))
        atomicAdd(&hist[(key >> shift) & 0xFFu], 1u);
    }
    __syncthreads();
    if (tid == 0) {
      unsigned krem = sh_krem, cum = 0u;
      int b = 255;
      for (; b > 0; --b) { if (cum + hist[b] >= krem) break; cum += hist[b]; }
      sh_prefix = prefix | ((unsigned)b << shift);
      sh_krem = krem - cum;
    }
    __syncthreads();
  }
  unsigned T = sh_prefix, krem = sh_krem;
  int ngt = k - (int)krem;
  if (tid == 0) { sh_cgt = 0; sh_ceq = 0; }
  __syncthreads();
  int* s = sel + (size_t)img * K_TOT + lvl * 1000;
  for (int i = tid; i < n; i += nth) {
    unsigned key = mono_key(ob[i]);
    if (key > T) {
      int p = atomicAdd(&sh_cgt, 1);
      if (p < ngt) s[p] = base + i;
    } else if (key == T) {
      int e = atomicAdd(&sh_ceq, 1);
      if (e < (int)krem) s[ngt + e] = base + i;
    }
  }
}

// ---------------- Kernel 2: gather + decode + sigmoid + clip + valid -------
__global__ void decode_kernel(const float* __restrict__ objn,
                              const float4* __restrict__ deltas,
                              const float4* __restrict__ anchors,
                              const int* __restrict__ sel,
                              float* __restrict__ dboxes, float* __restrict__ dscore,
                              float* __restrict__ dkey, int* __restrict__ dvalid,
                              int* __restrict__ dlvl, int* __restrict__ maxc) {
  int s = blockIdx.x * blockDim.x + threadIdx.x;
  if (s >= K_TOT) return;
  int img = blockIdx.y;
  int a = sel[(size_t)img * K_TOT + s];
  float4 an = anchors[a];
  float wa = an.z - an.x, ha = an.w - an.y;
  float cxa = an.x + 0.5f * wa, cya = an.y + 0.5f * ha;
  float4 d = deltas[(size_t)img * A_TOTAL + a];
  float dw = fminf(d.z, BBOX_CLAMP), dh = fminf(d.w, BBOX_CLAMP);
  float cx = d.x * wa + cxa, cy = d.y * ha + cya;
  float w = expf(dw) * wa, h = expf(dh) * ha;
  float x1 = fminf(fmaxf(cx - 0.5f * w, 0.0f), IMG_SZ);
  float y1 = fminf(fmaxf(cy - 0.5f * h, 0.0f), IMG_SZ);
  float x2 = fminf(fmaxf(cx + 0.5f * w, 0.0f), IMG_SZ);
  float y2 = fminf(fmaxf(cy + 0.5f * h, 0.0f), IMG_SZ);
  float obv = objn[(size_t)img * A_TOTAL + a];
  float sc = 1.0f / (1.0f + expf(-obv));
  int lvl = (a < 196608) ? 0 : (a < 245760) ? 1 : (a < 258048) ? 2 : (a < 261120) ? 3 : 4;
  int valid = ((x2 - x1) >= 1e-3f) && ((y2 - y1) >= 1e-3f) && (sc >= 0.0f);
  size_t idx = (size_t)img * K_TOT + s;
  dboxes[idx * 4 + 0] = x1; dboxes[idx * 4 + 1] = y1;
  dboxes[idx * 4 + 2] = x2; dboxes[idx * 4 + 3] = y2;
  dscore[idx] = sc;
  dkey[idx] = valid ? sc : -1.0f;
  dvalid[idx] = valid;
  dlvl[idx] = lvl;
  float m4 = fmaxf(fmaxf(x1, y1), fmaxf(x2, y2));
  atomicMax(&maxc[img], __float_as_int(m4));   // coords >= 0 -> int order == float order
}

// ---------------- Kernel 3: per-image bitonic sort (descending score) ------
__global__ void sort_kernel(const float* __restrict__ dkey, const float* __restrict__ dboxes,
                            const float* __restrict__ dscore, const int* __restrict__ dvalid,
                            const int* __restrict__ dlvl, const int* __restrict__ maxc,
                            float* __restrict__ sboxes, float* __restrict__ soffb,
                            float* __restrict__ sscore, int* __restrict__ svalid) {
  __shared__ unsigned long long keys[SORT_N];   // 64 KB
  int img = blockIdx.x, tid = threadIdx.x, nth = blockDim.x;
  for (int i = tid; i < SORT_N; i += nth) {
    unsigned long long kk = 0ull;
    if (i < K_TOT) {
      unsigned m = mono_key(dkey[(size_t)img * K_TOT + i]);
      kk = ((unsigned long long)m << 32) | (unsigned long long)(0xFFFFFFFFu - (unsigned)i);
    }
    keys[i] = kk;
  }
  __syncthreads();
  for (int k = 2; k <= SORT_N; k <<= 1)
    for (int j = k >> 1; j > 0; j >>= 1) {
      for (int i = tid; i < SORT_N; i += nth) {
        int ixj = i ^ j;
        if (ixj > i) {
          bool up = (i & k) == 0;
          unsigned long long va = keys[i], vb = keys[ixj];
          if ((va > vb) == up) { keys[i] = vb; keys[ixj] = va; }
        }
      }
      __syncthreads();
    }
  float mplus = __int_as_float(maxc[img]) + 1.0f;
  for (int r = tid; r < K_TOT; r += nth) {
    unsigned long long kk = keys[SORT_N - 1 - r];        // descending order
    int i = (int)(0xFFFFFFFFu - (unsigned)kk);
    size_t si = (size_t)img * K_TOT + i;
    size_t dd = (size_t)img * K_TOT + r;
    float off = (float)dlvl[si] * mplus;
    float b0 = dboxes[si * 4 + 0], b1 = dboxes[si * 4 + 1];
    float b2 = dboxes[si * 4 + 2], b3 = dboxes[si * 4 + 3];
    sboxes[dd * 4 + 0] = b0; sboxes[dd * 4 + 1] = b1;
    sboxes[dd * 4 + 2] = b2; sboxes[dd * 4 + 3] = b3;
    soffb[dd * 4 + 0] = b0 + off; soffb[dd * 4 + 1] = b1 + off;
    soffb[dd * 4 + 2] = b2 + off; soffb[dd * 4 + 3] = b3 + off;
    sscore[dd] = dscore[si];
    svalid[dd] = dvalid[si];
  }
}

// ---- Kernel 4: 16x16 IoU tiles. WMMA computes union base a_i + a_j; ------
// ---- i-boxes staged into LDS with gfx1250 async global->LDS copy. --------
__global__ void __launch_bounds__(32)
iou_mask_kernel(const float* __restrict__ soffb, unsigned* __restrict__ mask) {
  int tj = blockIdx.x, ti = blockIdx.y, img = blockIdx.z;
  if (tj < ti) return;                       // upper triangle only (bits where j > i)
  int lane = threadIdx.x;
  int col = lane & 15;
  __shared__ float4 tile[16];

  // async-copy the 16 i-boxes (16B each) into LDS (ASYNCcnt path)
  const float* ibase = soffb + ((size_t)img * K_TOT + (size_t)ti * 16) * 4;
  unsigned ldsa = (unsigned)(size_t)(&tile[col]);   // low 32 bits of flat LDS addr
  unsigned voff = (unsigned)(col * 16);
  asm volatile("global_load_async_to_lds_b128 %0, %1, %2"
               :: "v"(ldsa), "v"(voff), "s"(ibase) : "memory");
  asm volatile("s_wait_asynccnt 0" ::: "memory");

  int jg = tj * 16 + col;                    // K_TOT = 298*16, never OOB
  float4 bj = ((const float4*)soffb)[(size_t)img * K_TOT + jg];
  float area_j = (bj.z - bj.x) * (bj.w - bj.y);
  float4 rowb = tile[col];
  float area_i = (rowb.z - rowb.x) * (rowb.w - rowb.y);

  // D[i][j] = area_i*1 + 1*area_j via V_WMMA_F32_16X16X4_F32 (K padded with 0)
  v2f A, B;
  A.x = (lane < 16) ? area_i : 0.0f;   // A 16x4: lanes0-15 K=0,1 ; lanes16-31 K=2,3
  A.y = (lane < 16) ? 1.0f : 0.0f;
  B.x = (lane < 16) ? 1.0f : area_j;   // B 4x16: V0 lanes0-15 row K=0, lanes16-31 row K=1
  B.y = 0.0f;
  v8f acc = {};
  acc = __builtin_amdgcn_wmma_f32_16x16x4_f32(false, A, false, B, (short)0, acc,
                                              false, false);

  int half = lane >> 4;
#pragma unroll
  for (int r = 0; r < 8; ++r) {
    int row = r + 8 * half;                 // C/D layout: VGPR r holds M=r (lanes0-15), M=r+8
    float4 bi = tile[row];
    float ltx = fmaxf(bi.x, bj.x), lty = fmaxf(bi.y, bj.y);
    float rbx = fminf(bi.z, bj.z), rby = fminf(bi.w, bj.w);
    float iw = fmaxf(rbx - ltx, 0.0f), ih = fmaxf(rby - lty, 0.0f);
    float inter = iw * ih;
    float uni = acc[r] - inter;             // (a_i + a_j) - inter
    float iou = (uni > 0.0f) ? (inter / uni) : 0.0f;
    int ig = ti * 16 + row;
    bool pred = (iou > NMS_THR) && (jg > ig);
    unsigned ball = (unsigned)__ballot(pred);  // low16: rows M=r ; high16: rows M=r+8
    if (lane == 0) {
      unsigned w = (unsigned)(tj >> 1);
      unsigned sh = (unsigned)((tj & 1) * 16);
      atomicOr(&mask[((size_t)img * K_TOT + ti * 16 + r) * KW + w],
               (ball & 0xFFFFu) << sh);
      atomicOr(&mask[((size_t)img * K_TOT + ti * 16 + r + 8) * KW + w],
               ((ball >> 16) & 0xFFFFu) << sh);
    }
  }
}

// ---------------- Kernel 5: sequential greedy scan + compaction ------------
__global__ void nms_scan_kernel(const float4* __restrict__ sboxes,
                                const float* __restrict__ sscore,
                                const int* __restrict__ svalid,
                                const unsigned* __restrict__ mask,
                                float* __restrict__ out) {
  __shared__ unsigned removed[KW];
  __shared__ int s_keep, s_pos;
  int img = blockIdx.x, lane = threadIdx.x;
  for (int w = lane; w < KW; w += 32) removed[w] = 0u;
  if (lane == 0) s_pos = 0;
  __syncthreads();
  float* outb = out;                               // [16][1000][4]
  float* outs = out + (size_t)N_IMG * POST_NMS * 4; // [16][1000]
  for (int i = 0; i < K_TOT; ++i) {
    if (lane == 0) {
      bool rm = (removed[i >> 5] >> (i & 31)) & 1u;
      int kept = (svalid[(size_t)img * K_TOT + i] != 0) && !rm;
      s_keep = kept;
      if (kept) {
        int p = s_pos; s_pos = p + 1;
        if (p < POST_NMS) {
          float4 b = sboxes[(size_t)img * K_TOT + i];
          float* ob = outb + ((size_t)img * POST_NMS + p) * 4;
          ob[0] = b.x; ob[1] = b.y; ob[2] = b.z; ob[3] = b.w;
          outs[(size_t)img * POST_NMS + p] = sscore[(size_t)img * K_TOT + i];
        }
      }
    }
    __syncthreads();
    if (s_keep) {
      const unsigned* row = mask + ((size_t)img * K_TOT + i) * KW;
      for (int w = lane; w < KW; w += 32) removed[w] |= row[w];
    }
    __syncthreads();
  }
}

extern "C" void kernel_launch(void* const* d_in, const int* in_sizes, int n_in,
                              void* d_out, int out_size, void* d_ws, size_t ws_size,
                              hipStream_t stream) {
  const float* objn    = (const float*)d_in[0];   // [16, 261888]
  const float* deltas  = (const float*)d_in[1];   // [16, 261888, 4]
  const float* anchors = (const float*)d_in[2];   // [261888, 4]
  (void)in_sizes; (void)n_in;

  char* ws = (char*)d_ws;
  size_t o = 0;
  auto alloc = [&](size_t b) { size_t r = o; o += (b + 255) & ~(size_t)255; return r; };
  int*      sel    = (int*)     (ws + alloc(sizeof(int)   * N_IMG * K_TOT));
  float*    dboxes = (float*)   (ws + alloc(sizeof(float) * N_IMG * K_TOT * 4));
  float*    dscore = (float*)   (ws + alloc(sizeof(float) * N_IMG * K_TOT));
  float*    dkey   = (float*)   (ws + alloc(sizeof(float) * N_IMG * K_TOT));
  int*      dvalid = (int*)     (ws + alloc(sizeof(int)   * N_IMG * K_TOT));
  int*      dlvl   = (int*)     (ws + alloc(sizeof(int)   * N_IMG * K_TOT));
  int*      maxc   = (int*)     (ws + alloc(sizeof(int)   * N_IMG));
  float*    sboxes = (float*)   (ws + alloc(sizeof(float) * N_IMG * K_TOT * 4));
  float*    soffb  = (float*)   (ws + alloc(sizeof(float) * N_IMG * K_TOT * 4));
  float*    sscore = (float*)   (ws + alloc(sizeof(float) * N_IMG * K_TOT));
  int*      svalid = (int*)     (ws + alloc(sizeof(int)   * N_IMG * K_TOT));
  size_t mask_bytes = sizeof(unsigned) * (size_t)N_IMG * K_TOT * KW;   // ~45.5 MB
  unsigned* mask   = (unsigned*)(ws + alloc(mask_bytes));
  (void)ws_size;

  hipMemsetAsync(maxc, 0, sizeof(int) * N_IMG, stream);
  hipMemsetAsync(mask, 0, mask_bytes, stream);
  hipMemsetAsync(d_out, 0, (size_t)out_size * sizeof(float), stream);

  topk_kernel<<<dim3(N_IMG, 5), 256, 0, stream>>>(objn, sel);
  decode_kernel<<<dim3((K_TOT + 255) / 256, N_IMG), 256, 0, stream>>>(
      objn, (const float4*)deltas, (const float4*)anchors, sel,
      dboxes, dscore, dkey, dvalid, dlvl, maxc);
  sort_kernel<<<N_IMG, 1024, 0, stream>>>(dkey, dboxes, dscore, dvalid, dlvl,
                                          maxc, sboxes, soffb, sscore, svalid);
  iou_mask_kernel<<<dim3(NTILE, NTILE, N_IMG), 32, 0, stream>>>(soffb, mask);
  nms_scan_kernel<<<N_IMG, 32, 0, stream>>>((const float4*)sboxes, sscore, svalid,
                                            mask, (float*)d_out);
}